// wilson_9483287790378
// MI455X (gfx1250) — compile-verified
//
#include <hip/hip_runtime.h>

#define L 32
#define V (L*L*L*L)      // 1048576 sites
#define SY 32
#define SZ 1024
#define ST 32768
#define KOU 0.1f         // kappa / u0

struct C2 { float x, y; };

__device__ __forceinline__ C2 cadd(C2 a, C2 b) { return C2{a.x + b.x, a.y + b.y}; }

// acc + a*b
__device__ __forceinline__ C2 caccmul(C2 acc, C2 a, C2 b) {
    acc.x = fmaf(a.x, b.x, fmaf(-a.y, b.y, acc.x));
    acc.y = fmaf(a.x, b.y, fmaf( a.y, b.x, acc.y));
    return acc;
}
// acc + conj(a)*b
__device__ __forceinline__ C2 caccmulj(C2 acc, C2 a, C2 b) {
    acc.x = fmaf(a.x, b.x, fmaf( a.y, b.y, acc.x));
    acc.y = fmaf(a.x, b.y, fmaf(-a.y, b.x, acc.y));
    return acc;
}

// scale by constant factor: 0:+1  1:-1  2:+i  3:-i
template <int F>
__device__ __forceinline__ C2 cs(C2 v) {
    if constexpr (F == 0) return v;
    else if constexpr (F == 1) return C2{-v.x, -v.y};
    else if constexpr (F == 2) return C2{-v.y,  v.x};   // +i*v
    else                       return C2{ v.y, -v.x};   // -i*v
}

// One hopping term: project (h0 = psi0 + F0*psi[SB0], h1 = psi1 + F1*psi[SB1]),
// color-multiply by U (or U^dagger if DAG), reconstruct into acc.
// acc0 += g0; acc1 += g1; acc2 += R2*(G2?g1:g0); acc3 += R3*(G3?g1:g0)
template <int SB0, int F0, int SB1, int F1, int G2, int R2, int G3, int R3, bool DAG>
__device__ __forceinline__ void hop(const C2 psi[4][3], const C2 U[3][3], C2 acc[4][3]) {
    C2 h0[3], h1[3], g0[3], g1[3];
#pragma unroll
    for (int c = 0; c < 3; ++c) {
        h0[c] = cadd(psi[0][c], cs<F0>(psi[SB0][c]));
        h1[c] = cadd(psi[1][c], cs<F1>(psi[SB1][c]));
    }
#pragma unroll
    for (int Cc = 0; Cc < 3; ++Cc) {
        C2 a0{0.f, 0.f}, a1{0.f, 0.f};
#pragma unroll
        for (int c = 0; c < 3; ++c) {
            if constexpr (DAG) {   // g_C = sum_c conj(U[c][C]) h_c
                a0 = caccmulj(a0, U[c][Cc], h0[c]);
                a1 = caccmulj(a1, U[c][Cc], h1[c]);
            } else {               // g_C = sum_c U[C][c] h_c
                a0 = caccmul(a0, U[Cc][c], h0[c]);
                a1 = caccmul(a1, U[Cc][c], h1[c]);
            }
        }
        g0[Cc] = a0; g1[Cc] = a1;
    }
#pragma unroll
    for (int c = 0; c < 3; ++c) {
        acc[0][c] = cadd(acc[0][c], g0[c]);
        acc[1][c] = cadd(acc[1][c], g1[c]);
        acc[2][c] = cadd(acc[2][c], cs<R2>(G2 ? g1[c] : g0[c]));
        acc[3][c] = cadd(acc[3][c], cs<R3>(G3 ? g1[c] : g0[c]));
    }
}

__device__ __forceinline__ void load_spinor_g(const float* __restrict__ re,
                                              const float* __restrict__ im,
                                              int s, C2 psi[4][3]) {
#pragma unroll
    for (int sp = 0; sp < 4; ++sp)
#pragma unroll
        for (int c = 0; c < 3; ++c) {
            const int off = (sp * 3 + c) * V + s;
            psi[sp][c].x = re[off];
            psi[sp][c].y = im[off];
        }
}

// LDS tile layout: lds[sc(12)][reim(2)][yline(8)][x(32)]
__device__ __forceinline__ void load_spinor_l(const float* lds, int yl, int xx, C2 psi[4][3]) {
#pragma unroll
    for (int sp = 0; sp < 4; ++sp)
#pragma unroll
        for (int c = 0; c < 3; ++c) {
            const int sc = sp * 3 + c;
            psi[sp][c].x = lds[((sc * 2 + 0) * 8 + yl) * 32 + xx];
            psi[sp][c].y = lds[((sc * 2 + 1) * 8 + yl) * 32 + xx];
        }
}

__device__ __forceinline__ void load_U(const float* __restrict__ Ure,
                                       const float* __restrict__ Uim,
                                       int mu, int s, C2 U[3][3]) {
#pragma unroll
    for (int a = 0; a < 3; ++a)
#pragma unroll
        for (int b = 0; b < 3; ++b) {
            const int off = ((a * 3 + b) * 4 + mu) * V + s;
            U[a][b].x = Ure[off];
            U[a][b].y = Uim[off];
        }
}

// CDNA5 async global->LDS DMA (ASYNCcnt path). Inline asm: portable across
// ROCm 7.2 / amdgpu-toolchain builtin arity differences.
__device__ __forceinline__ void async_g2l_f32(const float* gp, const float* lp) {
    unsigned l32 = (unsigned)(unsigned long long)lp;  // generic LDS ptr low 32b == LDS offset
    unsigned long long g64 = (unsigned long long)gp;
    asm volatile("global_load_async_to_lds_b32 %0, %1, off"
                 :: "v"(l32), "v"(g64) : "memory");
}

__device__ __forceinline__ void wait_async0() {
#if __has_builtin(__builtin_amdgcn_s_wait_asynccnt)
    __builtin_amdgcn_s_wait_asynccnt(0);
#else
    asm volatile("s_wait_asynccnt 0" ::: "memory");
#endif
}

__global__ __launch_bounds__(256)
void wilson_dslash_kernel(const float* __restrict__ src_re,
                          const float* __restrict__ src_im,
                          const float* __restrict__ U_re,
                          const float* __restrict__ U_im,
                          float2* __restrict__ out) {
    __shared__ float lds[12 * 2 * 8 * 32];   // 24 KB: 8 x-lines of the center spinor

    const int x  = threadIdx.x & 31;         // wave32 lane == x coordinate
    const int yl = threadIdx.x >> 5;         // y-line within block (8 waves)
    const int b  = blockIdx.x;
    const int y  = ((b & 3) << 3) + yl;
    const int z  = (b >> 2) & 31;
    const int t  = (b >> 7);

    const int base = t * ST + z * SZ;
    const int site = base + y * SY + x;

    // ---- stage center spinors of this (32x8) tile into LDS via async DMA ----
#pragma unroll
    for (int sc = 0; sc < 12; ++sc) {
        async_g2l_f32(src_re + sc * V + site, &lds[((sc * 2 + 0) * 8 + yl) * 32 + x]);
        async_g2l_f32(src_im + sc * V + site, &lds[((sc * 2 + 1) * 8 + yl) * 32 + x]);
    }
    wait_async0();
    __syncthreads();

    const int xp = (x + 1) & 31, xm = (x + 31) & 31;
    const int yp = (y + 1) & 31, ym = (y + 31) & 31;
    const int zp = (z + 1) & 31, zm = (z + 31) & 31;
    const int tp = (t + 1) & 31, tm = (t + 31) & 31;

    const int site_xm = base + y * SY + xm;
    const int site_yp = base + yp * SY + x;
    const int site_ym = base + ym * SY + x;
    const int site_zp = t * ST + zp * SZ + y * SY + x;
    const int site_zm = t * ST + zm * SZ + y * SY + x;
    const int site_tp = tp * ST + z * SZ + y * SY + x;
    const int site_tm = tm * ST + z * SZ + y * SY + x;

    C2 acc[4][3];
#pragma unroll
    for (int sp = 0; sp < 4; ++sp)
#pragma unroll
        for (int c = 0; c < 3; ++c) acc[sp][c] = C2{0.f, 0.f};

    C2 psi[4][3];
    C2 Umat[3][3];

    // ---- mu = 0 (x): neighbors from LDS (periodic wrap inside the wave) ----
    load_spinor_l(lds, yl, xp, psi);
    load_U(U_re, U_im, 0, site, Umat);
    hop<3,3, 2,3, 1,2, 0,2, false>(psi, Umat, acc);   // (1-g0)

    load_spinor_l(lds, yl, xm, psi);
    load_U(U_re, U_im, 0, site_xm, Umat);
    hop<3,2, 2,2, 1,3, 0,3, true>(psi, Umat, acc);    // (1+g0)

    // ---- mu = 1 (y): interior lines from LDS, tile edges from global ----
    if (yl < 7) load_spinor_l(lds, yl + 1, x, psi);
    else        load_spinor_g(src_re, src_im, site_yp, psi);
    load_U(U_re, U_im, 1, site, Umat);
    hop<3,0, 2,1, 1,1, 0,0, false>(psi, Umat, acc);   // (1-g1)

    if (yl > 0) load_spinor_l(lds, yl - 1, x, psi);
    else        load_spinor_g(src_re, src_im, site_ym, psi);
    load_U(U_re, U_im, 1, site_ym, Umat);
    hop<3,1, 2,0, 1,0, 0,1, true>(psi, Umat, acc);    // (1+g1)

    // ---- mu = 2 (z): global (L2-resident) ----
    load_spinor_g(src_re, src_im, site_zp, psi);
    load_U(U_re, U_im, 2, site, Umat);
    hop<2,3, 3,2, 0,2, 1,3, false>(psi, Umat, acc);   // (1-g2)

    load_spinor_g(src_re, src_im, site_zm, psi);
    load_U(U_re, U_im, 2, site_zm, Umat);
    hop<2,2, 3,3, 0,3, 1,2, true>(psi, Umat, acc);    // (1+g2)

    // ---- mu = 3 (t): global (L2-resident) ----
    load_spinor_g(src_re, src_im, site_tp, psi);
    load_U(U_re, U_im, 3, site, Umat);
    hop<2,1, 3,1, 0,1, 1,1, false>(psi, Umat, acc);   // (1-g3)

    load_spinor_g(src_re, src_im, site_tm, psi);
    load_U(U_re, U_im, 3, site_tm, Umat);
    hop<2,0, 3,0, 0,0, 1,0, true>(psi, Umat, acc);    // (1+g3)

    // ---- dest = src - 0.1 * acc ; store interleaved complex64 as float2 ----
    C2 cen[4][3];
    load_spinor_l(lds, yl, x, cen);
#pragma unroll
    for (int sp = 0; sp < 4; ++sp)
#pragma unroll
        for (int c = 0; c < 3; ++c) {
            float2 o;
            o.x = fmaf(-KOU, acc[sp][c].x, cen[sp][c].x);
            o.y = fmaf(-KOU, acc[sp][c].y, cen[sp][c].y);
            out[(sp * 3 + c) * V + site] = o;
        }
}

extern "C" void kernel_launch(void* const* d_in, const int* in_sizes, int n_in,
                              void* d_out, int out_size, void* d_ws, size_t ws_size,
                              hipStream_t stream) {
    const float* src_re = (const float*)d_in[0];
    const float* src_im = (const float*)d_in[1];
    const float* U_re   = (const float*)d_in[2];
    const float* U_im   = (const float*)d_in[3];
    float2* out = (float2*)d_out;

    dim3 block(256, 1, 1);
    dim3 grid(32 * 32 * 4, 1, 1);   // (t,z) x 4 y-blocks of 8 lines
    wilson_dslash_kernel<<<grid, block, 0, stream>>>(src_re, src_im, U_re, U_im, out);
}